// MoEGate_10376640987565
// MI455X (gfx1250) — compile-verified
//
#include <hip/hip_runtime.h>
#include <hip/hip_bf16.h>

typedef __attribute__((ext_vector_type(2))) float v2f;
typedef __attribute__((ext_vector_type(4))) float v4f;
typedef __attribute__((ext_vector_type(8))) float v8f;

#define HDIM   2048
#define NEXP   16
#define ALPHA  0.01f
#define KCHUNK 512            // HDIM / 4 K-split chunks
#define TOK_PER_BLOCK 32      // 2 tiles x 16 tokens

// 8 waves/block: wave = (tile<<2) | kchunk. Each wave: 16x16 logits over K=512.
__global__ __launch_bounds__(256) void moe_gate_main(
    const float* __restrict__ x,      // [T, H]
    const float* __restrict__ w,      // [E, H]
    float* __restrict__ out,          // [2T idx | 2T weight | 1 aux]
    float* __restrict__ acc_ws,       // [16 Pi_sum | 16 cnt]
    int T) {
  __shared__ float w_lds[NEXP * HDIM];        // 128 KB: full gate weight
  __shared__ float partial_lds[8 * 16 * NEXP];// 8 KB: one 16x16 partial per wave
  __shared__ float blkPi[NEXP];
  __shared__ float blkCnt[NEXP];

  const int tid = threadIdx.x;

  // Cooperative load of the whole weight matrix into LDS (b128 loads).
  {
    const v4f* src = (const v4f*)w;
    v4f* dst = (v4f*)w_lds;
    #pragma unroll 8
    for (int i = 0; i < (NEXP * HDIM / 4) / 256; ++i)
      dst[tid + i * 256] = src[tid + i * 256];
  }
  if (tid < NEXP) { blkPi[tid] = 0.0f; blkCnt[tid] = 0.0f; }
  __syncthreads();

  const int wave   = tid >> 5;                 // 0..7
  const int lane   = tid & 31;
  const int tile   = wave >> 2;                // 0..1 (16-token tile in block)
  const int kchunk = wave & 3;                 // 0..3 (K-split)
  const int token0 = (blockIdx.x * 2 + tile) * 16;
  const int kbase  = kchunk * KCHUNK;

  const int mrow  = lane & 15;                 // token row (A) / expert (B)
  const int khalf = lane >> 4;                 // selects K pair {0,1} vs {2,3}

  // A fragment: x[token0+mrow][kbase + k + 2*khalf + {0,1}]  (global, b64)
  const float* aptr = x + (size_t)(token0 + mrow) * HDIM + kbase + 2 * khalf;
  // B fragment: w[mrow][kbase + k + 2*khalf + {0,1}]         (LDS, b64)
  const float* bptr = w_lds + mrow * HDIM + kbase + 2 * khalf;

  v8f acc = {};
  #pragma unroll 8
  for (int k = 0; k < KCHUNK; k += 8) {
    v2f a0 = *(const v2f*)(aptr + k);
    v2f a1 = *(const v2f*)(aptr + k + 4);
    v2f b0 = *(const v2f*)(bptr + k);
    v2f b1 = *(const v2f*)(bptr + k + 4);
    acc = __builtin_amdgcn_wmma_f32_16x16x4_f32(false, a0, false, b0,
                                                (short)0, acc, false, false);
    acc = __builtin_amdgcn_wmma_f32_16x16x4_f32(false, a1, false, b1,
                                                (short)0, acc, false, false);
  }

  // Spill partial D tile: lane l, vgpr r -> token (r + 8*khalf), expert (l&15)
  {
    float* prow = partial_lds + wave * 256;
    #pragma unroll
    for (int r = 0; r < 8; ++r)
      prow[(r + 8 * khalf) * NEXP + mrow] = acc[r];
  }
  __syncthreads();

  // First wave: deterministic K-chunk reduction + softmax + top2 (1 token/lane)
  if (tid < TOK_PER_BLOCK) {
    const int ptile = tid >> 4;                // which 16-token tile
    const int m     = tid & 15;                // token within tile
    const float* p  = partial_lds + (ptile * 4) * 256 + m * NEXP;

    float s[NEXP];
    float mx = -3.0e38f;
    #pragma unroll
    for (int e = 0; e < NEXP; ++e) {
      float v = (p[e] + p[256 + e]) + (p[512 + e] + p[768 + e]);
      s[e] = v;
      mx = fmaxf(mx, v);
    }
    float sum = 0.0f;
    #pragma unroll
    for (int e = 0; e < NEXP; ++e) { s[e] = __expf(s[e] - mx); sum += s[e]; }
    const float inv = 1.0f / sum;
    #pragma unroll
    for (int e = 0; e < NEXP; ++e) s[e] *= inv;

    // top-1 then top-2 (strict '>' => lowest index wins ties, like lax.top_k)
    int i1 = 0; float v1 = s[0];
    #pragma unroll
    for (int e = 1; e < NEXP; ++e) if (s[e] > v1) { v1 = s[e]; i1 = e; }
    int i2 = (i1 == 0) ? 1 : 0; float v2 = s[i2];
    #pragma unroll
    for (int e = 0; e < NEXP; ++e)
      if (e != i1 && s[e] > v2) { v2 = s[e]; i2 = e; }

    const int t = (blockIdx.x * 2 + ptile) * 16 + m;
    if (t < T) {
      out[(size_t)t * 2 + 0] = (float)i1;
      out[(size_t)t * 2 + 1] = (float)i2;
      float* outw = out + (size_t)2 * T;
      outw[(size_t)t * 2 + 0] = v1;
      outw[(size_t)t * 2 + 1] = v2;

      #pragma unroll
      for (int e = 0; e < NEXP; ++e) atomicAdd(&blkPi[e], s[e]);
      atomicAdd(&blkCnt[i1], 1.0f);
      atomicAdd(&blkCnt[i2], 1.0f);
    }
  }
  __syncthreads();

  if (tid < NEXP) {
    atomicAdd(&acc_ws[tid], blkPi[tid]);
    atomicAdd(&acc_ws[NEXP + tid], blkCnt[tid]);
  }
}

__global__ void moe_gate_init(float* acc_ws) {
  if (threadIdx.x < 2 * NEXP) acc_ws[threadIdx.x] = 0.0f;
}

__global__ void moe_gate_final(const float* __restrict__ acc_ws,
                               float* __restrict__ out, int T) {
  if (threadIdx.x == 0) {
    float aux = 0.0f;
    const float invT  = 1.0f / (float)T;
    const float invTK = 1.0f / (float)(T * 2);
    #pragma unroll
    for (int e = 0; e < NEXP; ++e) {
      float Pi = acc_ws[e] * invT;
      float fi = acc_ws[NEXP + e] * invTK * (float)NEXP;
      aux += Pi * fi;
    }
    out[(size_t)4 * T] = aux * ALPHA;
  }
}

extern "C" void kernel_launch(void* const* d_in, const int* in_sizes, int n_in,
                              void* d_out, int out_size, void* d_ws, size_t ws_size,
                              hipStream_t stream) {
  const float* x = (const float*)d_in[0];   // [4,4096,2048] fp32
  const float* w = (const float*)d_in[1];   // [16,2048] fp32
  float* out = (float*)d_out;               // 4T + 1 floats
  float* acc = (float*)d_ws;                // 32 floats of scratch

  const int T = in_sizes[0] / HDIM;         // 16384 tokens

  moe_gate_init<<<1, 64, 0, stream>>>(acc);
  const int blocks = (T + TOK_PER_BLOCK - 1) / TOK_PER_BLOCK;  // 512
  moe_gate_main<<<blocks, 256, 0, stream>>>(x, w, out, acc, T);
  moe_gate_final<<<1, 32, 0, stream>>>(acc, out, T);
}